// MoEGate_43928925503869
// MI455X (gfx1250) — compile-verified
//
#include <hip/hip_runtime.h>

typedef __attribute__((ext_vector_type(2))) float v2f;
typedef __attribute__((ext_vector_type(8))) float v8f;

#define HDIM 4096
#define NEXP 64
#define TOPK 8
#define NTOK 16384
#define WAVES_PER_BLOCK 8
#define TOK_PER_WAVE 16
#define TOK_PER_BLOCK (WAVES_PER_BLOCK * TOK_PER_WAVE) /* 128 */
#define LDS_ROW 65 /* pad to dodge LDS bank conflicts in top-k phase */

__global__ void moe_zero_counts(unsigned int* __restrict__ counts) {
    if (threadIdx.x < NEXP) counts[threadIdx.x] = 0u;
}

__global__ __launch_bounds__(256)
void moe_gate_kernel(const float* __restrict__ x,
                     const float* __restrict__ Wg,
                     float* __restrict__ scores_out,
                     float* __restrict__ idx_out,
                     unsigned int* __restrict__ g_counts) {
    __shared__ float lds_logits[TOK_PER_BLOCK * LDS_ROW];
    __shared__ unsigned int lds_counts[NEXP];

    const int tid   = threadIdx.x;
    const int wave  = tid >> 5;
    const int lane  = tid & 31;
    const int lhalf = lane >> 4;   // 0 or 1 (selects K pair / row group)
    const int lmod  = lane & 15;   // row (A) / column (B,C,D)

    if (tid < NEXP) lds_counts[tid] = 0u;

    const int tokenBase = blockIdx.x * TOK_PER_BLOCK + wave * TOK_PER_WAVE;

    // A-matrix (16x4 f32): lane holds row = lmod, K = 2*lhalf + {0,1}
    const float* pA  = x  + (size_t)(tokenBase + lmod) * HDIM + lhalf * 2;
    // B-matrix (4x16 f32): lane holds col = lmod, K = 2*lhalf + {0,1};
    // B column c of tile t is expert row (t*16 + c) of W_gate (W_gate^T).
    const float* pB0 = Wg + (size_t)( 0 + lmod) * HDIM + lhalf * 2;
    const float* pB1 = Wg + (size_t)(16 + lmod) * HDIM + lhalf * 2;
    const float* pB2 = Wg + (size_t)(32 + lmod) * HDIM + lhalf * 2;
    const float* pB3 = Wg + (size_t)(48 + lmod) * HDIM + lhalf * 2;

    v8f acc0 = {}, acc1 = {}, acc2 = {}, acc3 = {};

    #pragma unroll 4
    for (int k = 0; k < HDIM; k += 4) {
        v2f a  = __builtin_nontemporal_load((const v2f*)(pA + k)); // stream x, don't pollute L2
        v2f b0 = *(const v2f*)(pB0 + k);
        v2f b1 = *(const v2f*)(pB1 + k);
        v2f b2 = *(const v2f*)(pB2 + k);
        v2f b3 = *(const v2f*)(pB3 + k);
        acc0 = __builtin_amdgcn_wmma_f32_16x16x4_f32(false, a, false, b0, (short)0, acc0, false, false);
        acc1 = __builtin_amdgcn_wmma_f32_16x16x4_f32(false, a, false, b1, (short)0, acc1, false, false);
        acc2 = __builtin_amdgcn_wmma_f32_16x16x4_f32(false, a, false, b2, (short)0, acc2, false, false);
        acc3 = __builtin_amdgcn_wmma_f32_16x16x4_f32(false, a, false, b3, (short)0, acc3, false, false);
    }

    // C/D layout: VGPR i of lane L = row (i + 8*lhalf), col lmod. Spill to LDS.
    const int rowBase = wave * TOK_PER_WAVE + 8 * lhalf;
    #pragma unroll
    for (int i = 0; i < 8; ++i) {
        float* dst = &lds_logits[(rowBase + i) * LDS_ROW];
        dst[ 0 + lmod] = acc0[i];
        dst[16 + lmod] = acc1[i];
        dst[32 + lmod] = acc2[i];
        dst[48 + lmod] = acc3[i];
    }
    __syncthreads();

    // One thread per token: stable top-8 over 64 logits, softmax, usage counts.
    if (tid < TOK_PER_BLOCK) {
        const float* row = &lds_logits[tid * LDS_ROW];
        unsigned long long used = 0ull;
        float topv[TOPK];
        int   topi[TOPK];
        #pragma unroll
        for (int kk = 0; kk < TOPK; ++kk) {
            float best = -__builtin_inff();
            int   bi   = 0;
            for (int e = 0; e < NEXP; ++e) {
                float v = row[e];
                bool free_slot = (((used >> e) & 1ull) == 0ull);
                if (free_slot && v > best) { best = v; bi = e; } // strict > => lowest index on ties
            }
            used |= (1ull << bi);
            topv[kk] = best;
            topi[kk] = bi;
            atomicAdd(&lds_counts[bi], 1u);
        }
        // softmax over the 8 selected logits (sorted descending; max = topv[0])
        float m = topv[0];
        float ex[TOPK];
        float s = 0.0f;
        #pragma unroll
        for (int kk = 0; kk < TOPK; ++kk) { ex[kk] = __expf(topv[kk] - m); s += ex[kk]; }
        float inv = 1.0f / s;

        const int token = blockIdx.x * TOK_PER_BLOCK + tid;
        #pragma unroll
        for (int kk = 0; kk < TOPK; ++kk) {
            scores_out[(size_t)token * TOPK + kk] = ex[kk] * inv;
            idx_out[(size_t)token * TOPK + kk]    = (float)topi[kk];
        }
    }
    __syncthreads();

    if (tid < NEXP) {
        unsigned int c = lds_counts[tid];
        if (c) atomicAdd(&g_counts[tid], c);
    }
}

__global__ void moe_loss_kernel(const unsigned int* __restrict__ counts,
                                float* __restrict__ loss_out) {
    __shared__ float partial[NEXP];
    const int e = threadIdx.x;
    float load = (float)counts[e] * (1.0f / (float)NTOK);
    float d = load - (1.0f / 64.0f);
    partial[e] = d * d;
    __syncthreads();
    for (int s = 32; s > 0; s >>= 1) {
        if (e < s) partial[e] += partial[e + s];
        __syncthreads();
    }
    if (e == 0) loss_out[0] = 0.01f * partial[0];
}

extern "C" void kernel_launch(void* const* d_in, const int* in_sizes, int n_in,
                              void* d_out, int out_size, void* d_ws, size_t ws_size,
                              hipStream_t stream) {
    const float* x  = (const float*)d_in[0];   // [4,4096,4096] fp32
    const float* Wg = (const float*)d_in[1];   // [64,4096] fp32

    float* out    = (float*)d_out;
    float* scores = out;                                   // [16384,8]
    float* idxs   = out + (size_t)NTOK * TOPK;             // [16384,8] (indices as float)
    float* loss   = out + (size_t)2 * NTOK * TOPK;         // [1]

    unsigned int* counts = (unsigned int*)d_ws;            // 64 expert counters

    moe_zero_counts<<<1, 64, 0, stream>>>(counts);
    moe_gate_kernel<<<NTOK / TOK_PER_BLOCK, WAVES_PER_BLOCK * 32, 0, stream>>>(
        x, Wg, scores, idxs, counts);
    moe_loss_kernel<<<1, 64, 0, stream>>>(counts, loss);
}